// GCN_68564857913628
// MI455X (gfx1250) — compile-verified
//
#include <hip/hip_runtime.h>
#include <math.h>

// ---------------------------------------------------------------------------
// GATv2 x2 + mean-pool + linear head for MI455X (gfx1250, wave32, WMMA).
// Dense projections use V_WMMA_F32_16X16X4_F32 (fp32-exact). Edge softmax /
// aggregation is memory-bound scatter: 2 edge passes per layer with
// non-returning global float atomics (resolve in the 192MB L2).
// Weight staging into LDS uses GLOBAL_LOAD_ASYNC_TO_LDS_B128 when the
// toolchain declares the async builtins (ASYNCcnt path), else plain loads.
// ---------------------------------------------------------------------------

typedef __attribute__((ext_vector_type(2))) float v2f;
typedef __attribute__((ext_vector_type(8))) float v8f;
typedef __attribute__((ext_vector_type(4))) int   v4i;

#define NEG_SLOPE 0.2f
#define NUM_GRAPHS 256

#define AS1 __attribute__((address_space(1)))
#define AS3 __attribute__((address_space(3)))

#if __has_builtin(__builtin_amdgcn_global_load_async_to_lds_b128) && \
    __has_builtin(__builtin_amdgcn_s_wait_asynccnt)
#define HAVE_ASYNC_LDS 1
#else
#define HAVE_ASYNC_LDS 0
#endif

__device__ __forceinline__ v8f wmma_f32(v2f a, v2f b, v8f c) {
  // 8 args: (neg_a, A, neg_b, B, c_mod, C, reuse_a, reuse_b)
  return __builtin_amdgcn_wmma_f32_16x16x4_f32(
      false, a, false, b, (short)0, c, false, false);
}

// float atomic-max via signed/unsigned integer monotonicity (init = -inf)
__device__ __forceinline__ void atomicMaxF(float* addr, float v) {
  if (v >= 0.0f) atomicMax((int*)addr, __float_as_int(v));
  else           atomicMin((unsigned int*)addr, __float_as_uint(v));
}

__device__ __forceinline__ float lrelu(float f) {
  return f > 0.0f ? f : NEG_SLOPE * f;
}

// ---------------------------------------------------------------------------
__global__ void fill_k(float* __restrict__ p, long n, float v) {
  long i = (long)blockIdx.x * blockDim.x + threadIdx.x;
  if (i < n) p[i] = v;
}

// ---------------------------------------------------------------------------
// GEMM1: xlr[n,0:16] = x @ W1l, xlr[n,16:32] = x @ W1r.  x:[N,128] row-major.
// One wave per 16-row tile; K=128 -> 32 WMMA steps, dual accumulators.
// Weights staged in LDS (128x16 x2 = 16KB) via async-to-LDS when available.
__global__ void gemm1_k(const float* __restrict__ x,
                        const float* __restrict__ Wl,
                        const float* __restrict__ Wr,
                        float* __restrict__ xlr, int N) {
  __shared__ float lds[4096];
#if HAVE_ASYNC_LDS
  // 16B per lane per op, straight into LDS; flat addr[31:0] == LDS offset.
  for (unsigned i = threadIdx.x * 4; i < 4096; i += blockDim.x * 4) {
    const float* gsrc = (i < 2048) ? (Wl + i) : (Wr + (i - 2048));
    __builtin_amdgcn_global_load_async_to_lds_b128(
        (AS1 v4i*)(unsigned long long)(const void*)gsrc,
        (AS3 v4i*)(unsigned)(unsigned long long)(const void*)&lds[i],
        0, 0);
  }
  __builtin_amdgcn_s_wait_asynccnt(0);
  __syncthreads();
#else
  for (int i = threadIdx.x; i < 2048; i += blockDim.x) {
    lds[i]        = Wl[i];
    lds[2048 + i] = Wr[i];
  }
  __syncthreads();
#endif

  int wave = threadIdx.x >> 5;
  int lane = threadIdx.x & 31;
  int half = lane >> 4;     // 0: K pair {k,k+1}, 1: {k+2,k+3}  (ISA 16x4 A layout)
  int lm   = lane & 15;

  int nTiles = (N + 15) >> 4;
  int tile   = blockIdx.x * (blockDim.x >> 5) + wave;
  if (tile >= nTiles) return;                 // wave-uniform: EXEC stays all-1s
  int base = tile << 4;

  int arow = base + lm; if (arow > N - 1) arow = N - 1;   // clamp tail reads
  const float* xrow = x + (long)arow * 128;

  v8f cl = {}; v8f cr = {};
  for (int ks = 0; ks < 32; ++ks) {
    int k = ks * 4 + half * 2;
    v2f a;  a.x  = xrow[k];               a.y  = xrow[k + 1];
    v2f bl; bl.x = lds[k * 16 + lm];      bl.y = lds[(k + 1) * 16 + lm];
    v2f br; br.x = lds[2048 + k * 16 + lm];
            br.y = lds[2048 + (k + 1) * 16 + lm];
    cl = wmma_f32(a, bl, cl);
    cr = wmma_f32(a, br, cr);
  }
  // D layout: VGPR r -> rows r (lanes 0-15) / r+8 (lanes 16-31), col = lane&15
  #pragma unroll
  for (int r = 0; r < 8; ++r) {
    int row = base + r + half * 8;
    if (row < N) {
      xlr[(long)row * 32 + lm]      = cl[r];
      xlr[(long)row * 32 + 16 + lm] = cr[r];
    }
  }
}

// ---------------------------------------------------------------------------
// GEMM2: xlr2[n,0:8]=h@W2l, xlr2[n,8:16]=h@W2r, with B packed as one 16x16.
__global__ void gemm2_k(const float* __restrict__ h,
                        const float* __restrict__ Wl,
                        const float* __restrict__ Wr,
                        float* __restrict__ xlr2, int N) {
  __shared__ float lds[256];
  for (int i = threadIdx.x; i < 256; i += blockDim.x) {
    int k = i >> 4, n = i & 15;
    lds[i] = (n < 8) ? Wl[k * 8 + n] : Wr[k * 8 + (n - 8)];
  }
  __syncthreads();

  int wave = threadIdx.x >> 5;
  int lane = threadIdx.x & 31;
  int half = lane >> 4, lm = lane & 15;

  int nTiles = (N + 15) >> 4;
  int tile   = blockIdx.x * (blockDim.x >> 5) + wave;
  if (tile >= nTiles) return;
  int base = tile << 4;

  int arow = base + lm; if (arow > N - 1) arow = N - 1;
  const float* hrow = h + (long)arow * 16;

  v8f c = {};
  #pragma unroll
  for (int ks = 0; ks < 4; ++ks) {
    int k = ks * 4 + half * 2;
    v2f a; a.x = hrow[k];            a.y = hrow[k + 1];
    v2f b; b.x = lds[k * 16 + lm];   b.y = lds[(k + 1) * 16 + lm];
    c = wmma_f32(a, b, c);
  }
  #pragma unroll
  for (int r = 0; r < 8; ++r) {
    int row = base + r + half * 8;
    if (row < N) xlr2[(long)row * 16 + lm] = c[r];
  }
}

// ---------------------------------------------------------------------------
__device__ __forceinline__ void get_edge(const int* __restrict__ ei, int E,
                                         int t, int& s, int& d) {
  if (t < E) { s = ei[t]; d = ei[E + t]; }
  else       { s = t - E; d = t - E; }      // appended self-loops
}

// Layer-1 edge pass A: e = a1 . lrelu(xl[s] + xr[d]);  segment-max into m[d]
__global__ void edge1a_k(const int* __restrict__ ei, int E, int Etot,
                         const float* __restrict__ xlr,
                         const float* __restrict__ a1,
                         float* __restrict__ ebuf, float* __restrict__ m) {
  int t = blockIdx.x * blockDim.x + threadIdx.x;
  if (t >= Etot) return;
  if (t + 16384 < E) {                       // prefetch the index streams
    __builtin_prefetch(&ei[t + 16384], 0, 1);
    __builtin_prefetch(&ei[E + t + 16384], 0, 1);
  }
  int s, d; get_edge(ei, E, t, s, d);
  const float4* xl = (const float4*)(xlr + (long)s * 32);
  const float4* xr = (const float4*)(xlr + (long)d * 32 + 16);
  const float4* av = (const float4*)a1;
  float e = 0.0f;
  #pragma unroll
  for (int q = 0; q < 4; ++q) {
    float4 u = xl[q], v = xr[q], aa = av[q];
    e += aa.x * lrelu(u.x + v.x);
    e += aa.y * lrelu(u.y + v.y);
    e += aa.z * lrelu(u.z + v.z);
    e += aa.w * lrelu(u.w + v.w);
  }
  ebuf[t] = e;
  atomicMaxF(&m[d], e);
}

// Layer-1 edge pass B: p = exp(e - m[d]); denom[d]+=p; acc[d] += p*xl[s]
__global__ void edge1b_k(const int* __restrict__ ei, int E, int Etot,
                         const float* __restrict__ xlr,
                         const float* __restrict__ ebuf,
                         const float* __restrict__ m,
                         float* __restrict__ denom, float* __restrict__ acc) {
  int t = blockIdx.x * blockDim.x + threadIdx.x;
  if (t >= Etot) return;
  if (t + 16384 < E) {
    __builtin_prefetch(&ei[t + 16384], 0, 1);
    __builtin_prefetch(&ei[E + t + 16384], 0, 1);
  }
  int s, d; get_edge(ei, E, t, s, d);
  float mm = m[d]; if (mm < -3.0e38f) mm = 0.0f;   // isfinite guard
  float p = __expf(ebuf[t] - mm);
  atomicAdd(&denom[d], p);
  const float* xl = xlr + (long)s * 32;
  float* ap = acc + (long)d * 16;
  #pragma unroll
  for (int c = 0; c < 16; ++c) atomicAdd(&ap[c], p * xl[c]);
}

// h1 = relu(acc/denom + b1), in place
__global__ void node1_k(float* __restrict__ acc,
                        const float* __restrict__ denom,
                        const float* __restrict__ b, int N) {
  int t = blockIdx.x * blockDim.x + threadIdx.x;
  if (t >= N * 16) return;
  int n = t >> 4, c = t & 15;
  float v = acc[t] / fmaxf(denom[n], 1e-16f) + b[c];
  acc[t] = fmaxf(v, 0.0f);
}

// Layer-2 edge pass A (dim 8)
__global__ void edge2a_k(const int* __restrict__ ei, int E, int Etot,
                         const float* __restrict__ xlr2,
                         const float* __restrict__ a2,
                         float* __restrict__ ebuf, float* __restrict__ m) {
  int t = blockIdx.x * blockDim.x + threadIdx.x;
  if (t >= Etot) return;
  if (t + 16384 < E) {
    __builtin_prefetch(&ei[t + 16384], 0, 1);
    __builtin_prefetch(&ei[E + t + 16384], 0, 1);
  }
  int s, d; get_edge(ei, E, t, s, d);
  const float4* xl = (const float4*)(xlr2 + (long)s * 16);
  const float4* xr = (const float4*)(xlr2 + (long)d * 16 + 8);
  const float4* av = (const float4*)a2;
  float e = 0.0f;
  #pragma unroll
  for (int q = 0; q < 2; ++q) {
    float4 u = xl[q], v = xr[q], aa = av[q];
    e += aa.x * lrelu(u.x + v.x);
    e += aa.y * lrelu(u.y + v.y);
    e += aa.z * lrelu(u.z + v.z);
    e += aa.w * lrelu(u.w + v.w);
  }
  ebuf[t] = e;
  atomicMaxF(&m[d], e);
}

// Layer-2 edge pass B (dim 8)
__global__ void edge2b_k(const int* __restrict__ ei, int E, int Etot,
                         const float* __restrict__ xlr2,
                         const float* __restrict__ ebuf,
                         const float* __restrict__ m,
                         float* __restrict__ denom, float* __restrict__ acc) {
  int t = blockIdx.x * blockDim.x + threadIdx.x;
  if (t >= Etot) return;
  if (t + 16384 < E) {
    __builtin_prefetch(&ei[t + 16384], 0, 1);
    __builtin_prefetch(&ei[E + t + 16384], 0, 1);
  }
  int s, d; get_edge(ei, E, t, s, d);
  float mm = m[d]; if (mm < -3.0e38f) mm = 0.0f;
  float p = __expf(ebuf[t] - mm);
  atomicAdd(&denom[d], p);
  const float* xl = xlr2 + (long)s * 16;
  float* ap = acc + (long)d * 8;
  #pragma unroll
  for (int c = 0; c < 8; ++c) atomicAdd(&ap[c], p * xl[c]);
}

// h2 = relu(acc2/denom2 + b2) fused with graph mean-pool accumulation
__global__ void node2_pool_k(const float* __restrict__ acc2,
                             const float* __restrict__ denom2,
                             const float* __restrict__ b2,
                             const int* __restrict__ batch,
                             float* __restrict__ gsum, float* __restrict__ gcnt,
                             int N) {
  int n = blockIdx.x * blockDim.x + threadIdx.x;
  if (n >= N) return;
  float inv = 1.0f / fmaxf(denom2[n], 1e-16f);
  int g = batch[n];
  #pragma unroll
  for (int c = 0; c < 8; ++c) {
    float v = fmaxf(acc2[(long)n * 8 + c] * inv + b2[c], 0.0f);
    atomicAdd(&gsum[g * 8 + c], v);
  }
  atomicAdd(&gcnt[g], 1.0f);
}

__global__ void final_k(const float* __restrict__ gsum,
                        const float* __restrict__ gcnt,
                        const float* __restrict__ Wlin,
                        const float* __restrict__ blin,
                        float* __restrict__ out) {
  int g = threadIdx.x;
  if (g >= NUM_GRAPHS) return;
  float cnt = fmaxf(gcnt[g], 1.0f);
  float acc = blin[0];
  #pragma unroll
  for (int c = 0; c < 8; ++c) acc += (gsum[g * 8 + c] / cnt) * Wlin[c];
  out[g] = acc;
}

// ---------------------------------------------------------------------------
extern "C" void kernel_launch(void* const* d_in, const int* in_sizes, int n_in,
                              void* d_out, int out_size, void* d_ws, size_t ws_size,
                              hipStream_t stream) {
  const float* x    = (const float*)d_in[0];
  const int*   ei   = (const int*)  d_in[1];
  const int*   batch= (const int*)  d_in[2];
  const float* W1l  = (const float*)d_in[3];
  const float* W1r  = (const float*)d_in[4];
  const float* a1   = (const float*)d_in[5];
  const float* b1   = (const float*)d_in[6];
  const float* W2l  = (const float*)d_in[7];
  const float* W2r  = (const float*)d_in[8];
  const float* a2   = (const float*)d_in[9];
  const float* b2   = (const float*)d_in[10];
  const float* Wlin = (const float*)d_in[11];
  const float* blin = (const float*)d_in[12];
  float* out = (float*)d_out;

  const int N    = in_sizes[2];          // 100000
  const int E    = in_sizes[1] / 2;      // 3200000
  const int Etot = E + N;                // + self loops

  // ---- workspace layout (floats) ----
  float* ws = (float*)d_ws;
  long off = 0;
  float* xlr1   = ws + off; off += (long)N * 32;   // [N,32] = xl1 | xr1
  float* acc1   = ws + off; off += (long)N * 16;   // layer-1 accum -> h1
  float* denom1 = ws + off; off += N;
  float* m1     = ws + off; off += N;
  float* ebuf   = ws + off; off += Etot;           // reused by both layers
  float* xlr2   = ws + off; off += (long)N * 16;   // [N,16] = xl2 | xr2
  float* acc2   = ws + off; off += (long)N * 8;
  float* denom2 = ws + off; off += N;
  float* m2     = ws + off; off += N;
  float* gsum   = ws + off; off += NUM_GRAPHS * 8;
  float* gcnt   = ws + off; off += NUM_GRAPHS;
  (void)ws_size; (void)n_in; (void)out_size;

  const int T = 256;
  auto nb = [](long n, int t) { return (int)((n + t - 1) / t); };

  // deterministic re-init every call (graph replay safe)
  fill_k<<<nb((long)N * 17, T), T, 0, stream>>>(acc1, (long)N * 17, 0.0f); // acc1+denom1
  fill_k<<<nb(N, T), T, 0, stream>>>(m1, N, -INFINITY);
  fill_k<<<nb((long)N * 9, T), T, 0, stream>>>(acc2, (long)N * 9, 0.0f);  // acc2+denom2
  fill_k<<<nb(N, T), T, 0, stream>>>(m2, N, -INFINITY);
  fill_k<<<nb(NUM_GRAPHS * 9, T), T, 0, stream>>>(gsum, NUM_GRAPHS * 9, 0.0f);

  const int nTiles = (N + 15) / 16;
  const int gemmBlocks = (nTiles + 7) / 8;         // 8 waves (tiles) per block

  // ---- layer 1 ----
  gemm1_k<<<gemmBlocks, T, 0, stream>>>(x, W1l, W1r, xlr1, N);
  edge1a_k<<<nb(Etot, T), T, 0, stream>>>(ei, E, Etot, xlr1, a1, ebuf, m1);
  edge1b_k<<<nb(Etot, T), T, 0, stream>>>(ei, E, Etot, xlr1, ebuf, m1, denom1, acc1);
  node1_k<<<nb((long)N * 16, T), T, 0, stream>>>(acc1, denom1, b1, N);

  // ---- layer 2 ----
  gemm2_k<<<gemmBlocks, T, 0, stream>>>(acc1, W2l, W2r, xlr2, N);
  edge2a_k<<<nb(Etot, T), T, 0, stream>>>(ei, E, Etot, xlr2, a2, ebuf, m2);
  edge2b_k<<<nb(Etot, T), T, 0, stream>>>(ei, E, Etot, xlr2, ebuf, m2, denom2, acc2);

  // ---- pool + head ----
  node2_pool_k<<<nb(N, T), T, 0, stream>>>(acc2, denom2, b2, batch, gsum, gcnt, N);
  final_k<<<1, 256, 0, stream>>>(gsum, gcnt, Wlin, blin, out);
}